// TransducerJoint_59811714564218
// MI455X (gfx1250) — compile-verified
//
#include <hip/hip_runtime.h>
#include <hip/hip_bf16.h>
#include <math.h>

// Problem constants (from reference)
#define B_DIM 4
#define H1_DIM 512
#define H2_DIM 640
#define D_DIM 640
#define V1_DIM 1025
#define NTILES 65      // 65 * 16 = 1040 >= 1025
#define T_DIM 256
#define U_DIM 128

typedef __attribute__((ext_vector_type(2))) float v2f;
typedef __attribute__((ext_vector_type(8))) float v8f;

// ---------------------------------------------------------------------------
// Pre-swizzle W_out [640 x 1025] into WMMA B-fragment order:
//   fragment id = k4 * 65 + tile   (k4 = k/4, tile covers cols tile*16..+15)
//   within fragment: 32 lanes x float2 { W[kb][n], W[kb+1][n] },
//   kb = k4*4 + 2*(lane>>4), n = tile*16 + (lane&15); cols >= 1025 are zero.
// Total: 160*65*32 float2 = 2.66 MB.
// ---------------------------------------------------------------------------
__global__ void tj_swizzle_wout(const float* __restrict__ W, float* __restrict__ Wsw) {
    int idx = blockIdx.x * 256 + threadIdx.x;            // over 160*65*32
    if (idx >= (D_DIM / 4) * NTILES * 32) return;
    const int lane = idx & 31;
    const int rest = idx >> 5;
    const int tile = rest % NTILES;
    const int k4   = rest / NTILES;
    const int kb   = k4 * 4 + (lane >> 4) * 2;
    const int n    = tile * 16 + (lane & 15);
    float x = 0.0f, y = 0.0f;
    if (n < V1_DIM) {
        x = W[(size_t)kb * V1_DIM + n];
        y = W[(size_t)(kb + 1) * V1_DIM + n];
    }
    Wsw[(size_t)idx * 2]     = x;
    Wsw[(size_t)idx * 2 + 1] = y;
}

// ---------------------------------------------------------------------------
// Phase A: C[b][m][n] = sum_k A[b][k][m] * W[k][n] + bias[n]
// A is K-major (enc [B,H1,T] / pred [B,H2,U]) which matches the WMMA
// A-fragment layout directly. Compile-time M,N,K -> immediate addressing.
// Block: 256 threads = 8 waves arranged 2(M) x 4(N); WG tile = 32M x 64N.
// ---------------------------------------------------------------------------
template <int M, int N, int K>
__global__ __launch_bounds__(256) void tj_gemm_tn_bias(
    const float* __restrict__ A, const float* __restrict__ W,
    const float* __restrict__ bias, float* __restrict__ C) {
    const int b = blockIdx.z;
    const float* Ab = A + (size_t)b * K * M;
    float* Cb = C + (size_t)b * M * N;

    const int wave = __builtin_amdgcn_readfirstlane(threadIdx.x >> 5);
    const int lane = threadIdx.x & 31;
    const int i = wave >> 2, j = wave & 3;
    const int m0 = blockIdx.y * 32 + i * 16;
    const int n0 = blockIdx.x * 64 + j * 16;
    const int half = lane >> 4;
    const int l = lane & 15;

    v8f acc = {};
    for (int k = 0; k < K; k += 4) {
        const int ka = k + half * 2;
        v2f a, bf;
        a.x  = Ab[(size_t)(ka    ) * M + m0 + l];
        a.y  = Ab[(size_t)(ka + 1) * M + m0 + l];
        bf.x = W [(size_t)(ka    ) * N + n0 + l];
        bf.y = W [(size_t)(ka + 1) * N + n0 + l];
        acc = __builtin_amdgcn_wmma_f32_16x16x4_f32(false, a, false, bf,
                                                    (short)0, acc, false, false);
    }
#pragma unroll
    for (int jj = 0; jj < 8; ++jj) {
        const int row = m0 + jj + half * 8;
        const int col = n0 + l;
        Cb[(size_t)row * N + col] = acc[jj] + bias[col];
    }
}

// ---------------------------------------------------------------------------
// Phase B: fused  logits = relu(f+g) @ W_out + b_out  ->  log_softmax -> out
// One WG = 64 flattened (t,u) rows x all 1040 cols. 512 threads = 16 waves
// arranged 4(M-subtile) x 4(N-group). A strip in LDS (stride 644, conflict-
// free fragment reads). B fragments come pre-swizzled: one coalesced b64
// load per WMMA with immediate offsets (single address VGPR + s_clause).
// Softmax staged per 16-row subtile through a 16 x 1060 LDS slab, reduced
// with wave32 shuffles. Wave coordinates go through readfirstlane so all
// wave-level branches are scalar (EXEC stays all-ones around WMMA).
// ---------------------------------------------------------------------------
#define LDA 644
#define LDC 1060

__global__ __launch_bounds__(512, 1) void tj_joint_kernel(
    const float* __restrict__ f,    // [B,T,D]
    const float* __restrict__ g,    // [B,U,D]
    const float* __restrict__ Wsw,  // swizzled fragments
    const float* __restrict__ bo,   // [V1]
    float* __restrict__ out) {      // [B*T*U, V1]
    extern __shared__ float smem[];
    float* As = smem;                 // 64 x LDA
    float* Cs = smem + 64 * LDA;      // 16 x LDC

    const int r0 = blockIdx.x * 64;
    const int b  = r0 / (T_DIM * U_DIM);
    const int rr = r0 - b * (T_DIM * U_DIM);
    const int t  = rr / U_DIM;
    const int u0 = rr - t * U_DIM;    // strip never crosses t (128 % 64 == 0)

    const float* frow = f + (size_t)(b * T_DIM + t) * D_DIM;
    const float* grow = g + (size_t)(b * U_DIM + u0) * D_DIM;

    // Build A = relu(f_t + g_u) in LDS, coalesced over k.
    for (int idx = threadIdx.x; idx < 64 * D_DIM; idx += 512) {
        const int m = idx / D_DIM, k = idx - m * D_DIM;
        const float v = frow[k] + grow[(size_t)m * D_DIM + k];
        As[m * LDA + k] = v > 0.0f ? v : 0.0f;
    }
    __syncthreads();

    const int wave = __builtin_amdgcn_readfirstlane(threadIdx.x >> 5);  // scalar
    const int lane = threadIdx.x & 31;
    const int i = wave >> 2;          // M subtile 0..3   (scalar)
    const int j = wave & 3;           // N group  0..3    (scalar)
    const int half = lane >> 4;
    const int l = lane & 15;
    const int mrow = i * 16 + l;      // A row for this lane's fragment

    const v2f* wb = (const v2f*)Wsw + lane;   // single address VGPR; rest immediates

    v8f acc[16];
    v8f acc2 = {};
#pragma unroll
    for (int tt = 0; tt < 16; ++tt) acc[tt] = (v8f){};

    for (int k4 = 0; k4 < D_DIM / 4; ++k4) {
        const int ka = k4 * 4 + half * 2;
        const float* ap = &As[mrow * LDA + ka];
        v2f a;
        a.x = ap[0];
        a.y = ap[1];
        const size_t fb = (size_t)k4 * NTILES;
        // High-locality prefetch (WGP scope: 4 sibling waves share these
        // fragments through WGP$), two k-steps (~33 KB) ahead.
        if (k4 + 2 < D_DIM / 4)
            __builtin_prefetch((const void*)(wb + (fb + 2 * NTILES + j) * 32), 0, 3);
#pragma unroll
        for (int tt = 0; tt < 16; ++tt) {
            const v2f bf = wb[(fb + tt * 4 + j) * 32];
            acc[tt] = __builtin_amdgcn_wmma_f32_16x16x4_f32(false, a, false, bf,
                                                            (short)0, acc[tt],
                                                            false, false);
        }
        if (j == 0) {  // scalar branch: extra tile for cols 1024..1039
            const v2f bf = wb[(fb + 64) * 32];
            acc2 = __builtin_amdgcn_wmma_f32_16x16x4_f32(false, a, false, bf,
                                                         (short)0, acc2,
                                                         false, false);
        }
    }

    // Fused log_softmax: one 16-row subtile at a time through the LDS slab.
    for (int ii = 0; ii < 4; ++ii) {
        __syncthreads();              // readers of previous slab are done
        if (i == ii) {                // scalar branch
#pragma unroll
            for (int tt = 0; tt < 16; ++tt) {
                const int n0 = (tt * 4 + j) * 16;
#pragma unroll
                for (int jj = 0; jj < 8; ++jj)
                    Cs[(jj + half * 8) * LDC + n0 + l] = acc[tt][jj];
            }
            if (j == 0) {
#pragma unroll
                for (int jj = 0; jj < 8; ++jj)
                    Cs[(jj + half * 8) * LDC + 1024 + l] = acc2[jj];
            }
        }
        __syncthreads();              // slab complete

        const int row = wave;         // 16 waves <-> 16 rows
        float mx = -INFINITY;
        for (int c = lane; c < V1_DIM; c += 32)
            mx = fmaxf(mx, Cs[row * LDC + c] + bo[c]);
#pragma unroll
        for (int off = 16; off > 0; off >>= 1)
            mx = fmaxf(mx, __shfl_xor(mx, off, 32));

        float sum = 0.0f;
        for (int c = lane; c < V1_DIM; c += 32)
            sum += __expf(Cs[row * LDC + c] + bo[c] - mx);
#pragma unroll
        for (int off = 16; off > 0; off >>= 1)
            sum += __shfl_xor(sum, off, 32);

        const float lz = mx + __logf(sum);
        const size_t rbase = (size_t)(r0 + ii * 16 + row) * V1_DIM;
        for (int c = lane; c < V1_DIM; c += 32)
            out[rbase + c] = Cs[row * LDC + c] + bo[c] - lz;
    }
}

// ---------------------------------------------------------------------------
// Launcher
// ---------------------------------------------------------------------------
extern "C" void kernel_launch(void* const* d_in, const int* in_sizes, int n_in,
                              void* d_out, int out_size, void* d_ws, size_t ws_size,
                              hipStream_t stream) {
    const float* enc    = (const float*)d_in[0];  // [B,H1,T]
    const float* pred   = (const float*)d_in[1];  // [B,H2,U]
    const float* W_enc  = (const float*)d_in[2];  // [H1,D]
    const float* b_enc  = (const float*)d_in[3];  // [D]
    const float* W_pred = (const float*)d_in[4];  // [H2,D]
    const float* b_pred = (const float*)d_in[5];  // [D]
    const float* W_out  = (const float*)d_in[6];  // [D,V1]
    const float* b_out  = (const float*)d_in[7];  // [V1]
    float* out = (float*)d_out;

    // Workspace layout (~6.6 MB): f | g | Wsw
    float* f   = (float*)d_ws;                                  // B*T*D
    float* g   = f + (size_t)B_DIM * T_DIM * D_DIM;             // B*U*D
    float* Wsw = g + (size_t)B_DIM * U_DIM * D_DIM;             // 160*65*64

    const int nswz = (D_DIM / 4) * NTILES * 32;                 // 332800 threads
    tj_swizzle_wout<<<(nswz + 255) / 256, 256, 0, stream>>>(W_out, Wsw);

    // f[b,t,d] = enc^T @ W_enc + b_enc   (M=T, N=D, K=H1)
    tj_gemm_tn_bias<T_DIM, D_DIM, H1_DIM>
        <<<dim3(D_DIM / 64, T_DIM / 32, B_DIM), 256, 0, stream>>>(enc, W_enc, b_enc, f);
    // g[b,u,d] = pred^T @ W_pred + b_pred (M=U, N=D, K=H2)
    tj_gemm_tn_bias<U_DIM, D_DIM, H2_DIM>
        <<<dim3(D_DIM / 64, U_DIM / 32, B_DIM), 256, 0, stream>>>(pred, W_pred, b_pred, g);

    const size_t shmem = (size_t)(64 * LDA + 16 * LDC) * sizeof(float); // ~227 KB
    tj_joint_kernel<<<dim3((B_DIM * T_DIM * U_DIM) / 64), 512, shmem, stream>>>(
        f, g, Wsw, b_out, out);
}